// Conv2DCaps_76364518523207
// MI455X (gfx1250) — compile-verified
//
#include <hip/hip_runtime.h>
#include <hip/hip_bf16.h>

// ---------------------------------------------------------------------------
// Conv2DCaps dynamic routing for MI455X (gfx1250, wave32, WMMA).
//
// Shapes (fixed by reference): x (4,32,32,8,16), w (3,3,8,16,8,16)
// Output v: (4,30,30,8,16) fp32.
//
// u-GEMM runs on V_WMMA_F32_16X16X4_F32 (full fp32 precision, matches the
// fp32 reference exactly). u (151 MB fp32) is staged in d_ws; it fits in the
// 192 MB L2, so the 5 later passes over u are L2-resident.
// ---------------------------------------------------------------------------

typedef __attribute__((ext_vector_type(2))) float v2f;
typedef __attribute__((ext_vector_type(8))) float v8f;

namespace caps {
constexpr int B_  = 4,  H_ = 32, W_ = 32;
constexpr int CI  = 8,  NI = 16, CJ = 8, NJ = 16;
constexpr int KH  = 3,  KW = 3,  RC = KH * KW;          // 9
constexpr int HJ  = 30, WJ = 30;
constexpr int M_  = B_ * H_ * W_;                        // 4096 (b,h,w) rows
constexpr int ROWS = M_ * CI;                            // 32768 (b,hw,i) rows
constexpr int KKCJ = RC * CJ;                            // 72 routing columns
constexpr int JM   = CJ * NJ;                            // 128
constexpr long long U_ELEMS = (long long)ROWS * RC * JM; // 37,748,736
constexpr int  B_ELEMS = ROWS * KKCJ;                    // 2,359,296
constexpr int  R_ELEMS = ROWS;                           // 32,768
constexpr int  V_ELEMS = B_ * HJ * WJ * JM;              // 460,800
constexpr float EPS  = 1e-7f;
constexpr float JALL = (float)(HJ * WJ * CJ);            // 7200
constexpr float JADD = JALL - (float)KKCJ;               // 7128
}
using namespace caps;

// ---------------------------------------------------------------------------
// Kernel 0: zero the routing-logit state (b). d_ws is poisoned, not zeroed.
// ---------------------------------------------------------------------------
__global__ __launch_bounds__(256) void caps_zero(float* __restrict__ p, int n) {
  int idx = blockIdx.x * 256 + threadIdx.x;
  if (idx < n) p[idx] = 0.0f;
}

// ---------------------------------------------------------------------------
// Kernel 1: u[row, rc, j*16+m] = sum_n x[row_x, n] * w[rc, i, n, (j,m)]
// One wave per (m-tile, i, rc); each wave: 16x16 A tile in regs, sweeps the
// 128-wide (j,m) block as 8 N-tiles, 4 chained f32 WMMAs per tile (K=16).
// ISA layout (32-bit, wave32):
//   A 16x4 chunk: lanes 0-15 -> M=lane,  VGPR{0,1} = K{k0,k0+1}
//                 lanes 16-31-> M=lane-16, K{k0+2,k0+3}
//   B 4x16 chunk: lanes 0-15 -> N=lane,  VGPR{0,1} = rows k0,k0+1
//                 lanes 16-31-> N=lane-16, rows k0+2,k0+3
//   C/D: VGPR p  -> row M=p (lanes 0-15), row M=p+8 (lanes 16-31), N=lane&15
// ---------------------------------------------------------------------------
__global__ __launch_bounds__(256) void caps_u_gemm(const float* __restrict__ x,
                                                   const float* __restrict__ w,
                                                   float* __restrict__ u) {
  const int wid  = blockIdx.x * 8 + (threadIdx.x >> 5);   // 18432 waves total
  const int lane = threadIdx.x & 31;
  const int rc   = wid % RC;
  const int ci   = (wid / RC) % CI;
  const int mt   = wid / (RC * CI);                        // 0..255
  const int m0   = mt * 16;
  const int r16  = lane & 15;
  const int hi   = lane >> 4;

  // ---- A tile: row m of A_i is x[(m*CI+ci)*NI + 0..15] (contiguous 64B) ----
  const float* arow = x + (((m0 + r16) * CI) + ci) * NI + 2 * hi;
  v2f a[4];
  a[0] = *(const v2f*)(arow + 0);
  a[1] = *(const v2f*)(arow + 4);
  a[2] = *(const v2f*)(arow + 8);
  a[3] = *(const v2f*)(arow + 12);

  // ---- B block: B[n][col] = wb[n*128 + col], col in [0,128) for this rc ----
  const float* wb = w + (rc * CI + ci) * NI * JM;

  v8f acc[8];
  #pragma unroll
  for (int t = 0; t < 8; ++t) acc[t] = v8f{};

  #pragma unroll
  for (int t = 0; t < 8; ++t) {
    const int col = t * 16 + r16;
    #pragma unroll
    for (int k = 0; k < 4; ++k) {
      const int n0 = 4 * k + 2 * hi;
      v2f bb;
      bb.x = wb[(n0 + 0) * JM + col];
      bb.y = wb[(n0 + 1) * JM + col];
      acc[t] = __builtin_amdgcn_wmma_f32_16x16x4_f32(
          /*neg_a=*/false, a[k], /*neg_b=*/false, bb,
          /*c_mod=*/(short)0, acc[t], /*reuse_a=*/false, /*reuse_b=*/false);
    }
  }

  // ---- store D tiles: u[(row*RC + rc)*128 + col] ----
  #pragma unroll
  for (int t = 0; t < 8; ++t) {
    const int col = t * 16 + r16;
    #pragma unroll
    for (int p = 0; p < 8; ++p) {
      const int mrow = m0 + p + 8 * hi;
      u[(((long long)mrow * CI + ci) * RC + rc) * JM + col] = acc[t][p];
    }
  }
}

// ---------------------------------------------------------------------------
// Kernel 2: per (b,hw,i) row: recip = JALL / (sum_k exp(8*b[k]) + EPS + JADD)
// ---------------------------------------------------------------------------
__global__ __launch_bounds__(256) void caps_row_norm(const float* __restrict__ bst,
                                                     float* __restrict__ recip) {
  const int idx = blockIdx.x * 256 + threadIdx.x;
  if (idx >= ROWS) return;
  const float* bp = bst + (long long)idx * KKCJ;
  float s = 0.0f;
  #pragma unroll 8
  for (int k = 0; k < KKCJ; ++k) s += __expf(8.0f * bp[k]);
  recip[idx] = JALL / (s + EPS + JADD);
}

// ---------------------------------------------------------------------------
// Kernel 3: fused s-accumulate + squeeze -> v.
// One block per (b,hj,wj); thread t = j*16+m of the 128-wide (j,m) slab.
//   s[j,m] = sum_{r,c,i} exp(8*b[row,rcj]) * recip[row] * u[row,rc,j,m]
//   v = sq/(1+sq) * s / sqrt(sq+eps),  sq = sum_m s^2 (16-lane shuffle tree)
// ---------------------------------------------------------------------------
__global__ __launch_bounds__(128) void caps_s_v(const float* __restrict__ u,
                                                const float* __restrict__ bst,
                                                const float* __restrict__ recip,
                                                float* __restrict__ vout) {
  const int bid = blockIdx.x;                 // b*HJ*WJ + hj*WJ + wj
  const int b   = bid / (HJ * WJ);
  const int rem = bid % (HJ * WJ);
  const int hj  = rem / WJ, wj = rem % WJ;
  const int t   = threadIdx.x;                // j*16 + m
  const int j   = t >> 4;

  float s = 0.0f;
  #pragma unroll
  for (int r = 0; r < KH; ++r) {
    #pragma unroll
    for (int c = 0; c < KW; ++c) {
      const int hw    = (hj + r) * W_ + (wj + c);
      const int rcidx = r * KW + c;
      #pragma unroll
      for (int i = 0; i < CI; ++i) {
        const int rowi = (b * (H_ * W_) + hw) * CI + i;
        const float bv = bst[(long long)rowi * KKCJ + rcidx * CJ + j];
        const float cc = __expf(8.0f * bv) * recip[rowi];
        s += cc * u[((long long)rowi * RC + rcidx) * JM + t];
      }
    }
  }

  // 16-lane reduction of s^2 (masks 8/4/2/1 stay within the j-group)
  float sq = s * s;
  sq += __shfl_xor(sq, 8, 32);
  sq += __shfl_xor(sq, 4, 32);
  sq += __shfl_xor(sq, 2, 32);
  sq += __shfl_xor(sq, 1, 32);

  const float v = (sq / (1.0f + sq)) * s * rsqrtf(sq + EPS);
  vout[(long long)bid * JM + t] = v;
}

// ---------------------------------------------------------------------------
// Kernel 4: agreement + logit update.
// One thread per (b,hw,i,rcj): a = dot_m( v[b, h-r, w-c, j, :], u[row,rc,j,:] )
// b[idx] += a  (zero-padded v outside [0,HJ)x[0,WJ) -> no update).
// ---------------------------------------------------------------------------
__global__ __launch_bounds__(256) void caps_agree(const float* __restrict__ u,
                                                  const float* __restrict__ v,
                                                  float* __restrict__ bst) {
  const int idx  = blockIdx.x * 256 + threadIdx.x;   // rowi*72 + rcj (exact grid)
  const int rcj  = idx % KKCJ;
  const int rowi = idx / KKCJ;                       // (b*1024+hw)*8 + i
  const int hw   = (rowi >> 3) & (H_ * W_ - 1);
  const int b    = rowi >> 13;                       // / (1024*8)
  const int rc   = rcj >> 3, j = rcj & 7;
  const int r    = rc / KW, c = rc % KW;
  const int h    = hw >> 5, ww = hw & 31;
  const int hj   = h - r, wj = ww - c;

  if ((unsigned)hj < (unsigned)HJ && (unsigned)wj < (unsigned)WJ) {
    const float4* up = (const float4*)(u + ((long long)rowi * RC + rc) * JM + j * 16);
    const float4* vp = (const float4*)(v + ((long long)((b * HJ + hj) * WJ + wj)) * JM + j * 16);
    float acc = 0.0f;
    #pragma unroll
    for (int q = 0; q < 4; ++q) {
      const float4 uu = up[q];
      const float4 vv = vp[q];
      acc += uu.x * vv.x + uu.y * vv.y + uu.z * vv.z + uu.w * vv.w;
    }
    bst[idx] += acc;
  }
}

// ---------------------------------------------------------------------------
// Host-side orchestration (graph-capture safe: kernels only, one stream).
// Workspace layout (floats): u | bstate | recip | v  (~163 MB total).
// ---------------------------------------------------------------------------
extern "C" void kernel_launch(void* const* d_in, const int* in_sizes, int n_in,
                              void* d_out, int out_size, void* d_ws, size_t ws_size,
                              hipStream_t stream) {
  const float* x = (const float*)d_in[0];
  const float* w = (const float*)d_in[1];
  float* out = (float*)d_out;

  float* u_ws   = (float*)d_ws;
  float* bstate = u_ws + U_ELEMS;
  float* recip  = bstate + B_ELEMS;
  float* v_ws   = recip + R_ELEMS;

  // b-state <- 0 (ws is poisoned, never assume zeros)
  caps_zero<<<(B_ELEMS + 255) / 256, 256, 0, stream>>>(bstate, B_ELEMS);

  // u = einsum('bhwin,rcinjm->...') via fp32 WMMA GEMMs
  // waves = 256 mtiles * 8 i * 9 rc = 18432 -> 2304 blocks of 8 waves
  caps_u_gemm<<<2304, 256, 0, stream>>>(x, w, u_ws);

  for (int it = 0; it < 3; ++it) {
    caps_row_norm<<<(ROWS + 255) / 256, 256, 0, stream>>>(bstate, recip);
    float* vdst = (it == 2) ? out : v_ws;
    caps_s_v<<<B_ * HJ * WJ, 128, 0, stream>>>(u_ws, bstate, recip, vdst);
    if (it < 2) {
      caps_agree<<<B_ELEMS / 256, 256, 0, stream>>>(u_ws, v_ws, bstate);
    }
  }
}